// Quantizer_9019431321833
// MI455X (gfx1250) — compile-verified
//
#include <hip/hip_runtime.h>
#include <stdint.h>

// ---- CDNA5 vector types ----
typedef float v2f __attribute__((ext_vector_type(2)));
typedef float v8f __attribute__((ext_vector_type(8)));
typedef uint32_t v4u __attribute__((ext_vector_type(4)));
typedef int v8i __attribute__((ext_vector_type(8)));
typedef int v4i __attribute__((ext_vector_type(4)));

#ifndef __has_builtin
#define __has_builtin(x) 0
#endif

#if __has_builtin(__builtin_amdgcn_tensor_load_to_lds) && __has_builtin(__builtin_amdgcn_s_wait_tensorcnt)
#define USE_TDM 1
#else
#define USE_TDM 0
#endif

#if __has_builtin(__builtin_amdgcn_wmma_f32_16x16x4_f32)
#define USE_WMMA 1
#else
#define USE_WMMA 0
#endif

#define NCOL 4096
#define TPB 256        // 8 wave32
#define NWAVE 8
#define KPT 16         // elements per thread (16*256 = 4096)
#define ZLV 16         // zero-point candidates
#define MAXQF 15.0f
#define EPSQ 1e-8f

__global__ __launch_bounds__(TPB) void Quantizer_9019431321833_kernel(
    const float* __restrict__ x, const int* __restrict__ nump, float* __restrict__ out)
{
  __shared__ float s_x[NCOL];           // 16 KB row staging (TDM target)
  __shared__ float s_red[NWAVE * ZLV];  // per-wave, per-z partial scores

  const int tid  = threadIdx.x;
  const int lane = tid & 31;
  const int wave = tid >> 5;
  const int row  = blockIdx.x;
  const float* xrow = x + (size_t)row * NCOL;

  // ---------------- Stage row into LDS ----------------
#if USE_TDM
  if (wave == 0) {
    // Tensor DMA descriptor: 1 x 4096 f32 tile of a 1024 x 4096 tensor,
    // global_addr = tile (row) start, dest = s_x.
    uint64_t ga  = (uint64_t)(uintptr_t)xrow;
    uint32_t lds = (uint32_t)(uintptr_t)(&s_x[0]);   // LDS byte offset = flat addr[31:0]
    v4u g0;
    g0[0] = 1u;                                      // count=1 valid descriptor
    g0[1] = lds;                                     // lds_addr [63:32]
    g0[2] = (uint32_t)ga;                            // global_addr [95:64]
    g0[3] = (uint32_t)(ga >> 32) | (2u << 30);       // global_addr[56:32] | type=2 (image)
    v8i g1;
    g1[0] = 0x00020000;                              // wg_mask=0, data_size=2 (4B)
    g1[1] = 0x10000000;                              // tensor_dim0 = 4096 (low 16 -> bits 63:48)
    g1[2] = 0x04000000;                              // tensor_dim1 = 1024 (bits 111:80)
    g1[3] = 0x10000000;                              // tile_dim0 = 4096 (bits 127:112)
    g1[4] = 1;                                       // tile_dim1 = 1, tile_dim2 = 0
    g1[5] = 4096;                                    // tensor_dim0_stride (low 32)
    g1[6] = 0;
    g1[7] = 0;
    v4i z4; z4[0] = 0; z4[1] = 0; z4[2] = 0; z4[3] = 0;   // groups 2/3 unused (2D)
    v8i z8;
#pragma unroll
    for (int j = 0; j < 8; ++j) z8[j] = 0;
    __builtin_amdgcn_tensor_load_to_lds(g0, g1, z4, z4, z8, 0);
  }
  __builtin_amdgcn_s_wait_tensorcnt(0);
  __syncthreads();
#else
  for (int j = tid; j < NCOL; j += TPB) s_x[j] = xrow[j];
  __syncthreads();
#endif

  // Pull my 16 elements into VGPRs (stride-TPB mapping: conflict-free LDS,
  // perfectly coalesced final stores). Hot loop never touches memory again.
  float xv[KPT];
#pragma unroll
  for (int k = 0; k < KPT; ++k) xv[k] = s_x[k * TPB + tid];

  // ---------------- Row min / max ----------------
  float mn = xv[0], mx = xv[0];
#pragma unroll
  for (int k = 1; k < KPT; ++k) { mn = fminf(mn, xv[k]); mx = fmaxf(mx, xv[k]); }
#pragma unroll
  for (int m = 16; m >= 1; m >>= 1) {           // wave32 butterfly
    mn = fminf(mn, __shfl_xor(mn, m, 32));
    mx = fmaxf(mx, __shfl_xor(mx, m, 32));
  }
  if (lane == 0) { s_red[wave] = mn; s_red[NWAVE + wave] = mx; }
  __syncthreads();
  mn = s_red[0]; mx = s_red[NWAVE];
#pragma unroll
  for (int w = 1; w < NWAVE; ++w) {
    mn = fminf(mn, s_red[w]);
    mx = fmaxf(mx, s_red[NWAVE + w]);
  }
  __syncthreads();
  const float xmin = fminf(mn, 0.0f);
  const float xmax = fmaxf(mx, 0.0f);

  const int num = *nump;
  const float r_over_num = (xmax - xmin) / (float)num;

  float best_score = 1e10f;
  float best_min = xmin;
  float best_max = xmax;

  // ---------------- MSE grid search ----------------
  for (int i = 1; i <= num; ++i) {
    const float tmp_max = r_over_num * (float)i;
    const float scale = fmaxf(tmp_max * (1.0f / MAXQF), EPSQ);
    const float inv = 1.0f / scale;

    float acc[ZLV];
#pragma unroll
    for (int z = 0; z < ZLV; ++z) acc[z] = 0.0f;

#pragma unroll
    for (int k = 0; k < KPT; ++k) {
      const float xk = xv[k];
      const float xr = rintf(xk * inv);         // hoisted out of z-loop
#pragma unroll
      for (int z = 0; z < ZLV; ++z) {
        // q - zero == clip(xr, -z, 15-z); zero candidate z is exact
        const float c = fminf(fmaxf(xr, (float)(-z)), (float)(15 - z));
        const float d = fabsf(__builtin_fmaf(-scale, c, xk));
        acc[z] += __expf(2.4f * __logf(d));     // |d|^2.4 via v_log/v_exp
      }
    }

    // Wave-level allreduce of the 16 z-scores (wave32 butterfly).
#pragma unroll
    for (int m = 16; m >= 1; m >>= 1) {
#pragma unroll
      for (int z = 0; z < ZLV; ++z) acc[z] += __shfl_xor(acc[z], m, 32);
    }
    // Lane z (z<16) publishes this wave's sum for candidate z.
    float v = acc[0];
#pragma unroll
    for (int z = 1; z < ZLV; ++z) v = (lane == z) ? acc[z] : v;
    if (lane < 16) s_red[wave * ZLV + lane] = v;
    __syncthreads();

    float bs; int bz;
#if USE_WMMA
    {
      // Cross-wave reduce + broadcast with V_WMMA_F32_16X16X4_F32:
      // A(m=z, k=wave-within-quad) per documented 16x4 f32 layout,
      // B = all-ones (layout invariant)  =>  D(z, *) = sum over 8 waves.
      const bool hi = lane >= 16;
      const int mrow = lane & 15;
      v2f a0, a1, b;
      a0.x = s_red[(hi ? 2 : 0) * ZLV + mrow];   // A(m, k=0 / k=2)
      a0.y = s_red[(hi ? 3 : 1) * ZLV + mrow];   // A(m, k=1 / k=3)
      a1.x = s_red[(hi ? 6 : 4) * ZLV + mrow];
      a1.y = s_red[(hi ? 7 : 5) * ZLV + mrow];
      b.x = 1.0f; b.y = 1.0f;
      v8f c;
#pragma unroll
      for (int j = 0; j < 8; ++j) c[j] = 0.0f;
      c = __builtin_amdgcn_wmma_f32_16x16x4_f32(false, a0, false, b, (short)0, c, false, false);
      c = __builtin_amdgcn_wmma_f32_16x16x4_f32(false, a1, false, b, (short)0, c, false, false);
      // Lane holds D rows z = j + 8*hi; ascending-z scan keeps first minimum.
      bs = c[0] * (1.0f / (float)NCOL);
      bz = hi ? 8 : 0;
#pragma unroll
      for (int j = 1; j < 8; ++j) {
        const float s = c[j] * (1.0f / (float)NCOL);
        const int z = j + (hi ? 8 : 0);
        if (s < bs) { bs = s; bz = z; }
      }
      // Combine halves (lane ^ 16); ties -> smaller z (matches jnp.argmin).
      const float os = __shfl_xor(bs, 16, 32);
      const int   oz = __shfl_xor(bz, 16, 32);
      if (os < bs || (os == bs && oz < bz)) { bs = os; bz = oz; }
    }
#else
    {
      bs = 1e30f; bz = 0;
      for (int z = 0; z < ZLV; ++z) {
        float s = 0.0f;
        for (int w = 0; w < NWAVE; ++w) s += s_red[w * ZLV + z];
        s *= (1.0f / (float)NCOL);
        if (s < bs) { bs = s; bz = z; }
      }
    }
#endif
    if (bs < best_score) {
      best_min = -(float)bz * scale;            // new_min = -z*delta
      best_max = tmp_max - (float)bz * scale;   // new_max = tmp_max - z*delta
    }
    best_score = fminf(best_score, bs);
    __syncthreads();                            // protect s_red reuse next i
  }

  // ---------------- Final quantize ----------------
  const float mneg = fminf(best_min, 0.0f);
  const float mpos = fmaxf(best_max, 0.0f);
  const float sc = fmaxf((mpos - mneg) * (1.0f / MAXQF), EPSQ);
  const float zp = fminf(fmaxf(-rintf(mneg / sc), 0.0f), MAXQF);
  const float isc = 1.0f / sc;
  float* orow = out + (size_t)row * NCOL;
#pragma unroll
  for (int k = 0; k < KPT; ++k) {
    const float q = fminf(fmaxf(rintf(xv[k] * isc) + zp, 0.0f), MAXQF);
    orow[k * TPB + tid] = sc * (q - zp);        // coalesced
  }
}

extern "C" void kernel_launch(void* const* d_in, const int* in_sizes, int n_in,
                              void* d_out, int out_size, void* d_ws, size_t ws_size,
                              hipStream_t stream) {
  (void)n_in; (void)out_size; (void)d_ws; (void)ws_size;
  const float* x = (const float*)d_in[0];
  const int* num = (const int*)d_in[1];     // scalar 100 on device
  float* out = (float*)d_out;
  const int rows = in_sizes[0] / NCOL;      // 1024
  Quantizer_9019431321833_kernel<<<dim3(rows), dim3(TPB), 0, stream>>>(x, num, out);
}